// MixtureOfMamba_64484638982270
// MI455X (gfx1250) — compile-verified
//
#include <hip/hip_runtime.h>
#include <hip/hip_bf16.h>
#include <math.h>

// ---------------- problem constants (from reference) ----------------
#define DMODEL   1024
#define DSTATE   16
#define DCONV    4
#define DINNER   2048          // EXPAND * DMODEL
#define DTRANK   64            // ceil(1024/16)
#define NMOD     2
#define BSZ      2
#define SEQLEN   2048
#define TTOK     (BSZ * SEQLEN)   // 4096 tokens
#define XDBL_W   (DTRANK + 2 * DSTATE)  // 96

#define MTILES   4             // M-tiles per wave (64 rows x 16 cols per wave)

typedef __attribute__((ext_vector_type(2))) float v2f;
typedef __attribute__((ext_vector_type(8))) float v8f;

__device__ __forceinline__ float silu_f(float x) {
    return x / (1.0f + __expf(-x));
}

// =====================================================================
// Masked-expert WMMA GEMM (fp32, V_WMMA_F32_16X16X4_F32).
//   Out[t, o] = sum_k X[t, k] * W[id[t], o, k]   (+ bias / softplus)
// Implemented as   sum_e  (X .* [id==e]) @ W_e^T   — exact per-token
// expert selection via per-row masking of the A fragment.
// One wave computes a 64(M) x 16(N) tile. All dims are compile-time so
// the 4 A-subtile loads fold into immediate offsets off one base ptr;
// each K-step is a batched load burst followed by 8 back-to-back WMMAs.
// =====================================================================
template<int KDIM, int NDIM, int XLD, bool BIAS_SOFTPLUS>
__global__ __launch_bounds__(256) void gemm_moe_wmma(
    const float* __restrict__ X,     // (T, XLD) row-major, K values used per row
    const float* __restrict__ W,     // (NMOD, NDIM, KDIM) row-major
    const float* __restrict__ bias,  // (NMOD, NDIM) or nullptr
    const int*   __restrict__ ids,   // (T,)
    float*       __restrict__ Out)   // (T, NDIM)
{
    const int wave   = (blockIdx.x * blockDim.x + threadIdx.x) >> 5;
    const int lane   = threadIdx.x & 31;
    constexpr int ntiles = NDIM >> 4;
    const int tile_m = (wave / ntiles) << 6;       // 64 rows per wave
    const int tile_n = (wave % ntiles) << 4;

    const int laneHi = lane >> 4;                  // 0: K={0,1}, 1: K={2,3}
    const int lane16 = lane & 15;

    const int brow = tile_n + lane16;              // output row of W (B, N)
    const float* W0 = W + (size_t)brow * KDIM + laneHi;
    const float* W1 = W0 + (size_t)NDIM * KDIM;    // expert 1 weights

    // A base pointer; subtile i lives at constant offset i*16*XLD
    const float* Xp = X + (size_t)(tile_m + lane16) * XLD + (laneHi << 1);

    // per-token expert masks for each 16-row M sub-tile
    float m0[MTILES], m1[MTILES];
#pragma unroll
    for (int i = 0; i < MTILES; ++i) {
        m0[i] = (ids[tile_m + (i << 4) + lane16] == 0) ? 1.0f : 0.0f;
        m1[i] = 1.0f - m0[i];
    }

    v8f acc[MTILES];
#pragma unroll
    for (int i = 0; i < MTILES; ++i)
        acc[i] = (v8f){0.f, 0.f, 0.f, 0.f, 0.f, 0.f, 0.f, 0.f};

#pragma unroll 2
    for (int k = 0; k < KDIM; k += 4) {
        // ---- load phase: batched, one wait covers all ----
        v2f b0; b0.x = W0[k]; b0.y = W0[k + 2];    // expert-0 B fragment
        v2f b1; b1.x = W1[k]; b1.y = W1[k + 2];    // expert-1 B fragment
        v2f a[MTILES];
#pragma unroll
        for (int i = 0; i < MTILES; ++i)
            a[i] = *(const v2f*)(Xp + (size_t)i * 16 * XLD + k);

        // ---- compute phase: 8 WMMAs, 4 independent chains ----
#pragma unroll
        for (int i = 0; i < MTILES; ++i) {
            v2f a0 = a[i] * m0[i];
            acc[i] = __builtin_amdgcn_wmma_f32_16x16x4_f32(
                         false, a0, false, b0, (short)0, acc[i], false, false);
        }
#pragma unroll
        for (int i = 0; i < MTILES; ++i) {
            v2f a1 = a[i] * m1[i];
            acc[i] = __builtin_amdgcn_wmma_f32_16x16x4_f32(
                         false, a1, false, b1, (short)0, acc[i], false, false);
        }
    }

    // C/D layout: acc[i][v] -> row = tile_m + 16*i + v + 8*laneHi,
    //                          col = tile_n + lane16
#pragma unroll
    for (int i = 0; i < MTILES; ++i) {
        const int row0 = tile_m + (i << 4) + (laneHi << 3);
        size_t off = (size_t)row0 * NDIM + tile_n + lane16;
#pragma unroll
        for (int v = 0; v < 8; ++v) {
            float val = acc[i][v];
            if (BIAS_SOFTPLUS) {
                const int e = ids[row0 + v];
                val += bias[(size_t)e * NDIM + tile_n + lane16];
                val = (val > 20.0f) ? val : log1pf(__expf(val));  // softplus
            }
            Out[off] = val;
            off += NDIM;
        }
    }
}

// =====================================================================
// Depthwise causal conv (d_conv=4) + SiLU.  x is the first half of xz.
// =====================================================================
__global__ __launch_bounds__(256) void conv_silu_kernel(
    const float* __restrict__ xz,      // (T, 2*DINNER); x = [:, :DINNER]
    const float* __restrict__ conv_w,  // (DINNER, DCONV)
    const float* __restrict__ conv_b,  // (DINNER,)
    float*       __restrict__ xc)      // (T, DINNER)
{
    const int idx = blockIdx.x * blockDim.x + threadIdx.x;
    if (idx >= TTOK * DINNER) return;
    const int c = idx & (DINNER - 1);
    const int t = idx >> 11;                       // DINNER = 2048 = 2^11
    const int l = t & (SEQLEN - 1);                // SEQLEN = 2048 = 2^11

    float accv = conv_b[c];
#pragma unroll
    for (int k = 0; k < DCONV; ++k) {
        const int lp = l + k - (DCONV - 1);
        const float xv = (lp >= 0)
            ? xz[(size_t)(t + k - (DCONV - 1)) * (2 * DINNER) + c]
            : 0.0f;
        accv += conv_w[c * DCONV + k] * xv;
    }
    xc[(size_t)t * DINNER + c] = silu_f(accv);
}

// =====================================================================
// Selective scan, fused with D-skip and SiLU(z) gating.
// Wave32 = 2 channels x 16 states. lane%16 = state index n.
// =====================================================================
__global__ __launch_bounds__(256) void scan_fused_kernel(
    const float* __restrict__ xz,     // z = xz[:, DINNER:]
    const float* __restrict__ xc,     // (T, DINNER) post-conv activations
    const float* __restrict__ xdbl,   // (T, 96): [dt_raw | B | C]
    const float* __restrict__ dtb,    // (T, DINNER) softplus(dt)
    const float* __restrict__ A_log,  // (DINNER, DSTATE)
    const float* __restrict__ Dp,     // (DINNER,)
    float*       __restrict__ y)      // (T, DINNER)
{
    const int wave = (blockIdx.x * blockDim.x + threadIdx.x) >> 5;
    const int lane = threadIdx.x & 31;
    const int half = lane >> 4;
    const int n    = lane & 15;                    // state index

    const int wavesPerB = DINNER / 2;              // 2 channels per wave
    const int b = wave / wavesPerB;
    const int c = (wave % wavesPerB) * 2 + half;
    if (b >= BSZ) return;

    const float An = -__expf(A_log[c * DSTATE + n]);
    const float Dv = Dp[c];
    float state = 0.0f;

    for (int l = 0; l < SEQLEN; ++l) {
        const int t = b * SEQLEN + l;
        const float dtv = dtb[(size_t)t * DINNER + c];
        const float xv  = xc [(size_t)t * DINNER + c];
        const float Bv  = xdbl[(size_t)t * XDBL_W + DTRANK + n];
        const float Cv  = xdbl[(size_t)t * XDBL_W + DTRANK + DSTATE + n];

        state = state * __expf(dtv * An) + (dtv * xv) * Bv;

        float p = state * Cv;                      // reduce over 16 states
        p += __shfl_xor(p, 1, 32);
        p += __shfl_xor(p, 2, 32);
        p += __shfl_xor(p, 4, 32);
        p += __shfl_xor(p, 8, 32);

        if (n == 0) {
            const float zv = xz[(size_t)t * (2 * DINNER) + DINNER + c];
            y[(size_t)t * DINNER + c] = (p + xv * Dv) * silu_f(zv);
        }
    }
}

// =====================================================================
// Launch
// =====================================================================
extern "C" void kernel_launch(void* const* d_in, const int* in_sizes, int n_in,
                              void* d_out, int out_size, void* d_ws, size_t ws_size,
                              hipStream_t stream) {
    const float* hs     = (const float*)d_in[0];   // (B, L, DMODEL)
    const int*   ids    = (const int*)  d_in[1];   // (B, L)
    const float* W_in   = (const float*)d_in[2];   // (2, 2*DINNER, DMODEL)
    const float* conv_w = (const float*)d_in[3];   // (DINNER, DCONV)
    const float* conv_b = (const float*)d_in[4];   // (DINNER,)
    const float* W_x    = (const float*)d_in[5];   // (2, 96, DINNER)
    const float* W_dt   = (const float*)d_in[6];   // (2, DINNER, DTRANK)
    const float* b_dt   = (const float*)d_in[7];   // (2, DINNER)
    const float* A_log  = (const float*)d_in[8];   // (DINNER, DSTATE)
    const float* Dp     = (const float*)d_in[9];   // (DINNER,)
    const float* W_out  = (const float*)d_in[10];  // (2, DMODEL, DINNER)
    float*       out    = (float*)d_out;

    // workspace carve-up
    char* ws = (char*)d_ws;
    float* xz   = (float*)ws;                                   // T*4096
    float* xc   = (float*)(ws + (size_t)TTOK * 4096 * 4);       // T*2048
    float* xdbl = (float*)(ws + (size_t)TTOK * 6144 * 4);       // T*96
    float* dtb  = (float*)(ws + (size_t)TTOK * 6240 * 4);       // T*2048
    float* y    = (float*)(ws + (size_t)TTOK * 8288 * 4);       // T*2048

    const int threads = 256;

    // 1) in-projection: (T,1024) x (2,4096,1024) -> xz (T,4096)
    {
        int waves = (TTOK / 64) * ((2 * DINNER) / 16);          // 16384
        gemm_moe_wmma<DMODEL, 2 * DINNER, DMODEL, false>
            <<<waves * 32 / threads, threads, 0, stream>>>(
                hs, W_in, nullptr, ids, xz);
    }

    // 2) depthwise conv + SiLU -> xc (T,2048)
    {
        int total = TTOK * DINNER;
        conv_silu_kernel<<<(total + threads - 1) / threads, threads, 0, stream>>>(
            xz, conv_w, conv_b, xc);
    }

    // 3) x-projection: (T,2048) x (2,96,2048) -> xdbl (T,96)
    {
        int waves = (TTOK / 64) * (XDBL_W / 16);                // 384
        gemm_moe_wmma<DINNER, XDBL_W, DINNER, false>
            <<<waves * 32 / threads, threads, 0, stream>>>(
                xc, W_x, nullptr, ids, xdbl);
    }

    // 4) dt-projection + bias + softplus: (T,64 of 96) x (2,2048,64) -> dtb
    {
        int waves = (TTOK / 64) * (DINNER / 16);                // 8192
        gemm_moe_wmma<DTRANK, DINNER, XDBL_W, true>
            <<<waves * 32 / threads, threads, 0, stream>>>(
                xdbl, W_dt, b_dt, ids, dtb);
    }

    // 5) selective scan + D-skip + SiLU(z) gating -> y (T,2048)
    {
        int waves = BSZ * (DINNER / 2);                         // 2048
        scan_fused_kernel<<<waves * 32 / threads, threads, 0, stream>>>(
            xz, xc, xdbl, dtb, A_log, Dp, y);
    }

    // 6) out-projection: (T,2048) x (2,1024,2048) -> out (T,1024)
    {
        int waves = (TTOK / 64) * (DMODEL / 16);                // 4096
        gemm_moe_wmma<DINNER, DMODEL, DINNER, false>
            <<<waves * 32 / threads, threads, 0, stream>>>(
                y, W_out, nullptr, ids, out);
    }
}